// GCNLayer_21010980012326
// MI455X (gfx1250) — compile-verified
//
#include <hip/hip_runtime.h>

typedef __attribute__((ext_vector_type(2))) float v2f;
typedef __attribute__((ext_vector_type(8))) float v8f;

#define D_DIM 128
#define EDGES_PER_WAVE 4

// ---------------------------------------------------------------------------
// Kernel 1: zero the aggregation workspace (float4 stores, fully coalesced)
// ---------------------------------------------------------------------------
__global__ __launch_bounds__(256) void gcn_zero_kernel(float4* __restrict__ p,
                                                       long n4) {
  long i = (long)blockIdx.x * blockDim.x + threadIdx.x;
  if (i < n4) {
    p[i] = make_float4(0.f, 0.f, 0.f, 0.f);
  }
}

// ---------------------------------------------------------------------------
// Kernel 2: edge scatter.  One wave32 per 4 edges; lane l handles 4 consecutive
// floats of the 128-wide feature row.  x fits in L2 (51.2MB < 192MB) so the
// random gathers are L2-resident; atomics land on the L2 atomic units.
// ---------------------------------------------------------------------------
__global__ __launch_bounds__(256) void gcn_scatter_kernel(
    const float* __restrict__ x, const int* __restrict__ erow,
    const int* __restrict__ ecol, const float* __restrict__ eval,
    float* __restrict__ agg, int n_edges) {
  const int lane = threadIdx.x & 31;
  const int wave = threadIdx.x >> 5;
  const int waves_per_block = blockDim.x >> 5;
  const int e0 = (blockIdx.x * waves_per_block + wave) * EDGES_PER_WAVE;

  int rows[EDGES_PER_WAVE], cols[EDGES_PER_WAVE];
  float vals[EDGES_PER_WAVE];
#pragma unroll
  for (int j = 0; j < EDGES_PER_WAVE; ++j) {
    int e = e0 + j;
    if (e < n_edges) {
      rows[j] = erow[e];
      cols[j] = ecol[e];
      vals[j] = eval[e];
      // Pull the gather row toward the WGP before we need it
      // (lowers to global_prefetch_b8 on gfx1250).
      __builtin_prefetch(x + (size_t)cols[j] * D_DIM + lane * 4, 0, 0);
    } else {
      rows[j] = -1;
    }
  }

#pragma unroll
  for (int j = 0; j < EDGES_PER_WAVE; ++j) {
    if (rows[j] < 0) continue;
    const float4 m =
        *(const float4*)(x + (size_t)cols[j] * D_DIM + lane * 4);
    const float v = vals[j];
    float* dst = agg + (size_t)rows[j] * D_DIM + lane * 4;
    atomicAdd(dst + 0, v * m.x);
    atomicAdd(dst + 1, v * m.y);
    atomicAdd(dst + 2, v * m.z);
    atomicAdd(dst + 3, v * m.w);
  }
}

// ---------------------------------------------------------------------------
// Kernel 3: out = agg @ W + b via V_WMMA_F32_16X16X4_F32 (dtype-exact f32).
// Block = 256 threads = 8 waves.  Block owns a 16-row M tile; wave w owns the
// 16-column N tile w (8*16 = 128 = D_OUT).  K loop: 32 WMMAs of K=4.
//
// A fragment layout (ISA 7.12.2, 32-bit A 16x4): lanes 0-15 = rows, VGPR0/1
// hold K=k0,k0+1; lanes 16-31 hold K=k0+2,k0+3  -> one aligned float2 load.
// C/D layout: VGPR i, lanes 0-15 -> (M=i, N=lane), lanes 16-31 -> (M=i+8).
// Bias depends only on N, so the accumulator is seeded with b[col].
// ---------------------------------------------------------------------------
__global__ __launch_bounds__(256) void gcn_gemm_wmma_kernel(
    const float* __restrict__ agg, const float* __restrict__ W,
    const float* __restrict__ bias, float* __restrict__ out, int n_nodes) {
  const int lane = threadIdx.x & 31;
  const int wave = threadIdx.x >> 5;   // N tile index (0..7)
  const int mtile = blockIdx.x;        // M tile index
  const int half = lane >> 4;          // 0: lanes 0-15, 1: lanes 16-31
  const int l16 = lane & 15;

  const int row = (mtile << 4) + l16;            // A row held by this lane
  const int row_ld = row < n_nodes ? row : (n_nodes - 1);  // clamp tail loads
  const int col = (wave << 4) + l16;             // B/C/D column

  const float bval = bias[col];
  v8f acc;
#pragma unroll
  for (int i = 0; i < 8; ++i) acc[i] = bval;

  const float* arow = agg + (size_t)row_ld * D_DIM;

#pragma unroll
  for (int k0 = 0; k0 < D_DIM; k0 += 4) {
    const int ka = k0 + (half << 1);             // even -> 8B aligned
    v2f a = *(const v2f*)(arow + ka);            // A[row, ka], A[row, ka+1]
    v2f bb;
    bb.x = W[(size_t)ka * D_DIM + col];          // B[ka,   col]
    bb.y = W[(size_t)(ka + 1) * D_DIM + col];    // B[ka+1, col]
    acc = __builtin_amdgcn_wmma_f32_16x16x4_f32(
        /*neg_a=*/false, a, /*neg_b=*/false, bb,
        /*c_mod=*/(short)0, acc, /*reuse_a=*/false, /*reuse_b=*/false);
  }

  // Write D: VGPR i holds (M = mtile*16 + i + half*8, N = col)
  float* obase = out + (size_t)(mtile << 4) * D_DIM + col;
#pragma unroll
  for (int i = 0; i < 8; ++i) {
    const int m = (mtile << 4) + i + (half << 3);
    if (m < n_nodes) {
      obase[(size_t)(i + (half << 3)) * D_DIM] = acc[i];
    }
  }
}

// ---------------------------------------------------------------------------
// Launch
// ---------------------------------------------------------------------------
extern "C" void kernel_launch(void* const* d_in, const int* in_sizes, int n_in,
                              void* d_out, int out_size, void* d_ws,
                              size_t ws_size, hipStream_t stream) {
  const float* x = (const float*)d_in[0];     // [N, 128]
  const int* erow = (const int*)d_in[1];      // [E]
  const int* ecol = (const int*)d_in[2];      // [E]
  const float* eval = (const float*)d_in[3];  // [E]
  const float* W = (const float*)d_in[4];     // [128, 128]
  const float* b = (const float*)d_in[5];     // [128]
  float* out = (float*)d_out;                 // [N, 128]

  const int n_nodes = in_sizes[0] / D_DIM;
  const int n_edges = in_sizes[1];

  float* agg = (float*)d_ws;  // [N, 128] f32 scratch (51.2 MB for N=100000)

  // 1) zero agg
  const long n4 = (long)n_nodes * D_DIM / 4;
  const int zblocks = (int)((n4 + 255) / 256);
  gcn_zero_kernel<<<zblocks, 256, 0, stream>>>((float4*)agg, n4);

  // 2) scatter edges: 8 waves/block * EDGES_PER_WAVE edges/wave
  const int edges_per_block = 8 * EDGES_PER_WAVE;
  const int sblocks = (n_edges + edges_per_block - 1) / edges_per_block;
  gcn_scatter_kernel<<<sblocks, 256, 0, stream>>>(x, erow, ecol, eval, agg,
                                                  n_edges);

  // 3) WMMA GEMM + bias: one block per 16-row tile
  const int mtiles = (n_nodes + 15) / 16;
  gcn_gemm_wmma_kernel<<<mtiles, 256, 0, stream>>>(agg, W, b, out, n_nodes);
}